// GraphNet_57604101374099
// MI455X (gfx1250) — compile-verified
//
#include <hip/hip_runtime.h>
#include <hip/hip_bf16.h>

#define N_NODES   100000
#define E_EDGES   1600000
#define G_GRAPHS  64
#define IN_DIM    3
#define H_DIM     64
#define F1_DIM    256
#define OUT_DIM   10
#define BN_EPS    1e-5f
#define NSTRIPS   (N_NODES / 16)   // 6250, exact

typedef float v2f __attribute__((ext_vector_type(2)));
typedef float v8f __attribute__((ext_vector_type(8)));

// ---------------------------------------------------------------------------
// Edge scatter for layer 1: agg1[dst][c] += x[src][c], c in 0..2
// ---------------------------------------------------------------------------
__global__ void scatter1_kernel(const float* __restrict__ x,
                                const int* __restrict__ src,
                                const int* __restrict__ dst,
                                float* __restrict__ agg1) {
    int e = blockIdx.x * blockDim.x + threadIdx.x;
    if (e >= E_EDGES) return;
    int s = src[e], d = dst[e];
    atomicAdd(&agg1[d * 3 + 0], x[s * 3 + 0]);
    atomicAdd(&agg1[d * 3 + 1], x[s * 3 + 1]);
    atomicAdd(&agg1[d * 3 + 2], x[s * 3 + 2]);
}

// ---------------------------------------------------------------------------
// Layer 1: h1 = relu(agg1 @ W_rel1 + b_rel1 + x @ W_root1), IN=3 so scalar FMA
// ---------------------------------------------------------------------------
__global__ void layer1_kernel(const float* __restrict__ x,
                              const float* __restrict__ agg1,
                              const float* __restrict__ Wrel,
                              const float* __restrict__ brel,
                              const float* __restrict__ Wroot,
                              float* __restrict__ h1) {
    int idx = blockIdx.x * blockDim.x + threadIdx.x;   // N*64 threads
    if (idx >= N_NODES * H_DIM) return;
    int n = idx >> 6;
    int j = idx & 63;
    const float* xa = x + n * 3;
    const float* aa = agg1 + n * 3;
    float acc = brel[j];
#pragma unroll
    for (int c = 0; c < IN_DIM; ++c)
        acc += aa[c] * Wrel[c * H_DIM + j] + xa[c] * Wroot[c * H_DIM + j];
    h1[idx] = fmaxf(acc, 0.0f);
}

// ---------------------------------------------------------------------------
// Per-channel sum / sum-of-squares over rows (C divides 256)
// ---------------------------------------------------------------------------
__global__ void channel_stats_kernel(const float* __restrict__ h,
                                     float* __restrict__ sum,
                                     float* __restrict__ sumsq,
                                     long total, int C) {
    __shared__ float s_sum[256];
    __shared__ float s_sq[256];
    int t = threadIdx.x;
    for (int i = t; i < C; i += blockDim.x) { s_sum[i] = 0.0f; s_sq[i] = 0.0f; }
    __syncthreads();
    long idx    = (long)blockIdx.x * blockDim.x + t;
    long stride = (long)gridDim.x * blockDim.x;       // multiple of C
    int  ch     = (int)(idx % C);
    float ps = 0.0f, pq = 0.0f;
    for (; idx < total; idx += stride) {
        float v = h[idx];
        ps += v; pq += v * v;
    }
    atomicAdd(&s_sum[ch], ps);
    atomicAdd(&s_sq[ch], pq);
    __syncthreads();
    for (int i = t; i < C; i += blockDim.x) {
        atomicAdd(&sum[i],   s_sum[i]);
        atomicAdd(&sumsq[i], s_sq[i]);
    }
}

// ---------------------------------------------------------------------------
// In-place BatchNorm (training mode, biased variance)
// ---------------------------------------------------------------------------
__global__ void bn_apply_kernel(float* __restrict__ h,
                                const float* __restrict__ sum,
                                const float* __restrict__ sumsq,
                                const float* __restrict__ g,
                                const float* __restrict__ b,
                                long total, int C, float inv_n) {
    long idx = (long)blockIdx.x * blockDim.x + threadIdx.x;
    if (idx >= total) return;
    int ch  = (int)(idx % C);
    float m = sum[ch] * inv_n;
    float v = fmaxf(sumsq[ch] * inv_n - m * m, 0.0f);
    float sc = g[ch] * rsqrtf(v + BN_EPS);
    h[idx] = (h[idx] - m) * sc + b[ch];
}

// ---------------------------------------------------------------------------
// Edge scatter for layer 2: agg[dst][0..63] += h1[src][0..63]
// One thread per (edge, 4-channel chunk); float4 gather + 4 float atomics.
// Working set (h1 + agg = 51 MB) is L2-resident on MI455X (192 MB L2).
// ---------------------------------------------------------------------------
__global__ void scatter2_kernel(const float* __restrict__ h,
                                const int* __restrict__ src,
                                const int* __restrict__ dst,
                                float* __restrict__ agg) {
    long idx = (long)blockIdx.x * blockDim.x + threadIdx.x;  // E*16 threads
    int e = (int)(idx >> 4);
    if (e >= E_EDGES) return;
    int q = ((int)idx & 15) * 4;
    int s = src[e], d = dst[e];
    const float4 v = *(const float4*)(h + (size_t)s * H_DIM + q);
    float* p = agg + (size_t)d * H_DIM + q;
    atomicAdd(p + 0, v.x);
    atomicAdd(p + 1, v.y);
    atomicAdd(p + 2, v.z);
    atomicAdd(p + 3, v.w);
}

// ---------------------------------------------------------------------------
// Layer 2 GEMM via fp32 WMMA:
//   out = relu(agg @ W_rel2 + b_rel2 + h1 @ W_root2), written IN PLACE to agg.
// One wave per 16-row strip; 4 col-tiles of 16; K=64 in steps of 4.
// A layout (16x4 f32): lane%16 = row M; VGPR i holds K = k0 + i + 2*(lane/16).
// B layout (4x16 f32): lane%16 = col N; VGPR i holds K = k0 + i + 2*(lane/16).
// D layout (16x16 f32): VGPR r, lane -> row r + 8*(lane/16), col lane%16.
// ---------------------------------------------------------------------------
__global__ __launch_bounds__(256) void gemm2_wmma_kernel(
        float* agg,                          // A_rel source AND output (in-place)
        const float* __restrict__ h1,        // A_root source
        const float* __restrict__ Wrel,
        const float* __restrict__ Wroot,
        const float* __restrict__ bias) {
    __shared__ float sWrel[H_DIM * H_DIM];
    __shared__ float sWroot[H_DIM * H_DIM];
    __shared__ float sBias[H_DIM];

    int t = threadIdx.x;
    for (int i = t; i < H_DIM * H_DIM; i += 256) {
        sWrel[i]  = Wrel[i];
        sWroot[i] = Wroot[i];
    }
    if (t < H_DIM) sBias[t] = bias[t];
    __syncthreads();

    int wave  = t >> 5;
    int lane  = t & 31;
    int strip = blockIdx.x * 8 + wave;
    if (strip >= NSTRIPS) return;

    int row0 = strip * 16;
    int lr   = lane & 15;   // A row / B,D col within tile
    int hi   = lane >> 4;   // 0 or 1 -> K offset +2 for A/B, row offset +8 for D

    const float* Arel  = agg + (size_t)(row0 + lr) * H_DIM;
    const float* Aroot = h1  + (size_t)(row0 + lr) * H_DIM;

    v8f acc[4];
#pragma unroll
    for (int nt = 0; nt < 4; ++nt) acc[nt] = (v8f){};

#pragma unroll
    for (int k0 = 0; k0 < H_DIM; k0 += 4) {
        const int ka = k0 + 2 * hi;
        v2f a_rel  = *(const v2f*)(Arel  + ka);
        v2f a_root = *(const v2f*)(Aroot + ka);
#pragma unroll
        for (int nt = 0; nt < 4; ++nt) {
            const int col = nt * 16 + lr;
            v2f br;
            br.x = sWrel[ka * H_DIM + col];
            br.y = sWrel[(ka + 1) * H_DIM + col];
            acc[nt] = __builtin_amdgcn_wmma_f32_16x16x4_f32(
                false, a_rel, false, br, (short)0, acc[nt], false, false);
            v2f bo;
            bo.x = sWroot[ka * H_DIM + col];
            bo.y = sWroot[(ka + 1) * H_DIM + col];
            acc[nt] = __builtin_amdgcn_wmma_f32_16x16x4_f32(
                false, a_root, false, bo, (short)0, acc[nt], false, false);
        }
    }

    // bias + relu epilogue, in-place store (this wave has finished reading its rows)
#pragma unroll
    for (int nt = 0; nt < 4; ++nt) {
        const int col = nt * 16 + lr;
        const float bj = sBias[col];
#pragma unroll
        for (int r = 0; r < 8; ++r) {
            int m = row0 + r + 8 * hi;
            agg[(size_t)m * H_DIM + col] = fmaxf(acc[nt][r] + bj, 0.0f);
        }
    }
}

// ---------------------------------------------------------------------------
// Global mean pool: pooled[g] += h2[n]; cnt[g] += 1  (divide in fc1)
// ---------------------------------------------------------------------------
__global__ void pool_kernel(const float* __restrict__ h,
                            const int* __restrict__ membership,
                            float* __restrict__ pooled,
                            float* __restrict__ cnt) {
    long idx = (long)blockIdx.x * blockDim.x + threadIdx.x;  // N*64
    if (idx >= (long)N_NODES * H_DIM) return;
    int n = (int)(idx >> 6);
    int j = (int)(idx & 63);
    int g = membership[n];
    atomicAdd(&pooled[g * H_DIM + j], h[idx]);
    if (j == 0) atomicAdd(&cnt[g], 1.0f);
}

// ---------------------------------------------------------------------------
// FC1: z = relu(mean_pooled @ W_fc1 + b_fc1)   (64x64 @ 64x256)
// ---------------------------------------------------------------------------
__global__ void fc1_kernel(const float* __restrict__ pooled,
                           const float* __restrict__ cnt,
                           const float* __restrict__ W,
                           const float* __restrict__ b,
                           float* __restrict__ z) {
    int g = blockIdx.x;          // 64 blocks
    int f = threadIdx.x;         // 256 threads
    __shared__ float sp[H_DIM];
    if (f < H_DIM) sp[f] = pooled[g * H_DIM + f] / fmaxf(cnt[g], 1.0f);
    __syncthreads();
    float acc = b[f];
#pragma unroll 8
    for (int k = 0; k < H_DIM; ++k) acc += sp[k] * W[k * F1_DIM + f];
    z[g * F1_DIM + f] = fmaxf(acc, 0.0f);
}

// ---------------------------------------------------------------------------
// FC2: out = z_bn @ W_fc2 + b_fc2   (64x256 @ 256x10)
// ---------------------------------------------------------------------------
__global__ void fc2_kernel(const float* __restrict__ z,
                           const float* __restrict__ W,
                           const float* __restrict__ b,
                           float* __restrict__ out) {
    int g = blockIdx.x;          // 64 blocks
    int o = threadIdx.x;         // 16 threads, 10 active
    if (o >= OUT_DIM) return;
    float acc = b[o];
    for (int f = 0; f < F1_DIM; ++f) acc += z[g * F1_DIM + f] * W[f * OUT_DIM + o];
    out[g * OUT_DIM + o] = acc;
}

// ---------------------------------------------------------------------------
extern "C" void kernel_launch(void* const* d_in, const int* in_sizes, int n_in,
                              void* d_out, int out_size, void* d_ws, size_t ws_size,
                              hipStream_t stream) {
    (void)in_sizes; (void)n_in; (void)out_size; (void)ws_size;

    const float* x        = (const float*)d_in[0];
    const int*   member   = (const int*)  d_in[1];
    const int*   edges    = (const int*)  d_in[2];
    const int*   src      = edges;
    const int*   dst      = edges + E_EDGES;
    const float* W_rel1   = (const float*)d_in[3];
    const float* b_rel1   = (const float*)d_in[4];
    const float* W_root1  = (const float*)d_in[5];
    const float* W_rel2   = (const float*)d_in[6];
    const float* b_rel2   = (const float*)d_in[7];
    const float* W_root2  = (const float*)d_in[8];
    const float* g1       = (const float*)d_in[9];
    const float* b1       = (const float*)d_in[10];
    const float* g2       = (const float*)d_in[11];
    const float* b2       = (const float*)d_in[12];
    const float* W_fc1    = (const float*)d_in[13];
    const float* b_fc1    = (const float*)d_in[14];
    const float* g3       = (const float*)d_in[15];
    const float* b3       = (const float*)d_in[16];
    const float* W_fc2    = (const float*)d_in[17];
    const float* b_fc2    = (const float*)d_in[18];
    float*       out      = (float*)d_out;

    // Workspace layout (floats):
    //   H1   : N*64                      (layer-1 output, BN'd in place)
    //   AGG  : N*64                      (agg1 in first N*3, then agg2, then h2 in place)
    //   STATS: 21312                     (sums, cnt, pooled, z)
    float* ws     = (float*)d_ws;
    float* H1     = ws;
    float* AGG    = ws + (size_t)N_NODES * H_DIM;
    float* STATS  = AGG + (size_t)N_NODES * H_DIM;
    float* sum1   = STATS + 0;
    float* sumsq1 = STATS + 64;
    float* sum2   = STATS + 128;
    float* sumsq2 = STATS + 192;
    float* sumfc  = STATS + 256;   // 256
    float* sumsqfc= STATS + 512;   // 256
    float* cnt    = STATS + 768;   // 64
    float* pooled = STATS + 832;   // 64*64 = 4096
    float* z      = STATS + 4928;  // 64*256 = 16384  -> total 21312 floats

    const long NH = (long)N_NODES * H_DIM;         // 6,400,000 (divisible by 256)

    // ---- Layer 1 -----------------------------------------------------------
    hipMemsetAsync(AGG,   0, (size_t)NH * sizeof(float), stream);
    hipMemsetAsync(STATS, 0, 21312 * sizeof(float),      stream);

    scatter1_kernel<<<E_EDGES / 256, 256, 0, stream>>>(x, src, dst, AGG);
    layer1_kernel<<<(int)(NH / 256), 256, 0, stream>>>(x, AGG, W_rel1, b_rel1, W_root1, H1);
    channel_stats_kernel<<<256, 256, 0, stream>>>(H1, sum1, sumsq1, NH, H_DIM);
    bn_apply_kernel<<<(int)(NH / 256), 256, 0, stream>>>(H1, sum1, sumsq1, g1, b1,
                                                         NH, H_DIM, 1.0f / N_NODES);

    // ---- Layer 2 -----------------------------------------------------------
    hipMemsetAsync(AGG, 0, (size_t)NH * sizeof(float), stream);
    scatter2_kernel<<<(E_EDGES * 16) / 256, 256, 0, stream>>>(H1, src, dst, AGG);
    gemm2_wmma_kernel<<<(NSTRIPS + 7) / 8, 256, 0, stream>>>(AGG, H1, W_rel2, W_root2, b_rel2);
    channel_stats_kernel<<<256, 256, 0, stream>>>(AGG, sum2, sumsq2, NH, H_DIM);
    bn_apply_kernel<<<(int)(NH / 256), 256, 0, stream>>>(AGG, sum2, sumsq2, g2, b2,
                                                         NH, H_DIM, 1.0f / N_NODES);

    // ---- Pool + MLP head ---------------------------------------------------
    pool_kernel<<<(int)(NH / 256), 256, 0, stream>>>(AGG, member, pooled, cnt);
    fc1_kernel<<<G_GRAPHS, F1_DIM, 0, stream>>>(pooled, cnt, W_fc1, b_fc1, z);
    channel_stats_kernel<<<64, 256, 0, stream>>>(z, sumfc, sumsqfc,
                                                 (long)G_GRAPHS * F1_DIM, F1_DIM);
    bn_apply_kernel<<<(G_GRAPHS * F1_DIM) / 256, 256, 0, stream>>>(
        z, sumfc, sumsqfc, g3, b3, (long)G_GRAPHS * F1_DIM, F1_DIM, 1.0f / G_GRAPHS);
    fc2_kernel<<<G_GRAPHS, 16, 0, stream>>>(z, W_fc2, b_fc2, out);
}